// StochasticTwoLayerRGAT_35648228557574
// MI455X (gfx1250) — compile-verified
//
#include <hip/hip_runtime.h>
#include <hip/hip_bf16.h>
#include <math.h>

typedef float v2f __attribute__((ext_vector_type(2)));
typedef float v8f __attribute__((ext_vector_type(8)));
typedef unsigned int u32x4 __attribute__((ext_vector_type(4)));
typedef int i32x4 __attribute__((ext_vector_type(4)));
typedef int i32x8 __attribute__((ext_vector_type(8)));

#define NEG_SLOPE 0.2f

#if __has_builtin(__builtin_amdgcn_tensor_load_to_lds) && \
    __has_builtin(__builtin_amdgcn_s_wait_tensorcnt)
#define HAVE_TDM 1
#else
#define HAVE_TDM 0
#endif

// ---------------------------------------------------------------------------
// WMMA fp32 GEMM: C[N,128] = A[N,128] @ B[128,128]
// Block = 4 waves; the block stages B[0:128, by*64 : by*64+64] (32 KB) into
// LDS once (via the Tensor Data Mover when the builtin is available), then
// each wave computes one 16x16 tile of C with V_WMMA_F32_16X16X4_F32,
// accumulating K=128 in steps of 4. grid = (N/16, 2); N=50000 % 16 == 0.
// ---------------------------------------------------------------------------
__global__ __launch_bounds__(128)
void gemm128_wmma_f32(const float* __restrict__ A,
                      const float* __restrict__ B,
                      float* __restrict__ C, int N) {
    __shared__ float Blds[128 * 64];             // [k][c], c in 0..63
    const int lane = threadIdx.x & 31;
    const int wave = threadIdx.x >> 5;           // 0..3
    const int tm   = blockIdx.x * 16;            // row tile base
    const int by   = blockIdx.y;                 // column half: cols [by*64, +64)
    const int half = lane >> 4;                  // 0 or 1
    const int l    = lane & 15;

    // ---- stage B tile into LDS ----
#if HAVE_TDM
    if (threadIdx.x == 0) {
        unsigned long long gaddr = (unsigned long long)(const void*)(B + by * 64);
        unsigned lds_off = (unsigned)(unsigned long long)(const void*)&Blds[0];
        // D# group 0: count=1 user descriptor, lds addr, global addr, type=2
        u32x4 g0;
        g0[0] = 1u;
        g0[1] = lds_off;
        g0[2] = (unsigned)(gaddr & 0xffffffffull);
        g0[3] = (unsigned)((gaddr >> 32) & 0x01ffffffull) | (2u << 30);
        // D# group 1: data_size=4B, tensor 128x128, tile 64(X) x 128(Y),
        // row stride 128 elements, no multicast / barrier / padding.
        const unsigned dim0 = 128u, dim1 = 128u, t0 = 64u, t1 = 128u;
        const unsigned long long s0 = 128ull;
        i32x8 g1;
        g1[0] = (int)(2u << 16);                                  // data_size=4B
        g1[1] = (int)((dim0 & 0xffffu) << 16);                    // dim0[15:0]
        g1[2] = (int)(((dim0 >> 16) & 0xffffu) | ((dim1 & 0xffffu) << 16));
        g1[3] = (int)(((dim1 >> 16) & 0xffffu) | ((t0 & 0xffffu) << 16));
        g1[4] = (int)(t1 & 0xffffu);                              // tile_dim1
        g1[5] = (int)(s0 & 0xffffffffull);                        // stride0 lo
        g1[6] = (int)((s0 >> 32) & 0xffffull);                    // stride0 hi
        g1[7] = 0;
        i32x4 g2 = {0, 0, 0, 0};
        i32x4 g3 = {0, 0, 0, 0};
#if defined(__clang_major__) && (__clang_major__ >= 23)
        i32x8 g4 = {0, 0, 0, 0, 0, 0, 0, 0};
        __builtin_amdgcn_tensor_load_to_lds(g0, g1, g2, g3, g4, 0);
#else
        __builtin_amdgcn_tensor_load_to_lds(g0, g1, g2, g3, 0);
#endif
        __builtin_amdgcn_s_wait_tensorcnt(0);
    }
    __syncthreads();
#else
    // cooperative fallback: 128 rows x 16 float4 each
    for (int idx = threadIdx.x; idx < 128 * 16; idx += 128) {
        int k = idx >> 4, c4 = idx & 15;
        *(float4*)&Blds[k * 64 + c4 * 4] =
            *(const float4*)(B + k * 128 + by * 64 + c4 * 4);
    }
    __syncthreads();
#endif

    const int arow = tm + l;                     // A row for this lane
    const int cc   = wave * 16 + l;              // column within LDS tile

    v8f c = {};
#pragma unroll
    for (int k0 = 0; k0 < 128; k0 += 4) {
        const int k = k0 + half * 2;             // even -> 8B aligned
        // A 16x4 tile: lane<16 holds K=k0..k0+1, lane>=16 holds K=k0+2..k0+3
        v2f a = *(const v2f*)(A + arow * 128 + k);
        // B 4x16 tile from LDS (lanes read consecutive banks)
        v2f b;
        b.x = Blds[(k + 0) * 64 + cc];
        b.y = Blds[(k + 1) * 64 + cc];
        c = __builtin_amdgcn_wmma_f32_16x16x4_f32(
                /*neg_a=*/false, a, /*neg_b=*/false, b,
                /*c_mod=*/(short)0, c, /*reuse_a=*/false, /*reuse_b=*/false);
    }

    const int bcol = by * 64 + cc;
    const int crow = tm + half * 8;
#pragma unroll
    for (int v = 0; v < 8; ++v)
        C[(crow + v) * 128 + bcol] = c[v];
}

// ---------------------------------------------------------------------------
// Per-(node,head) attention score dots: el = <feat[n,h,:],al[h,:]>, er likewise
// ---------------------------------------------------------------------------
__global__ void score_kernel(const float* __restrict__ feat,
                             const float* __restrict__ al,
                             const float* __restrict__ ar,
                             float* __restrict__ el,
                             float* __restrict__ er, int N) {
    int i = blockIdx.x * blockDim.x + threadIdx.x;   // over N*4
    if (i >= N * 4) return;
    int n = i >> 2, h = i & 3;
    const float* f  = feat + n * 128 + h * 32;
    const float* wl = al + h * 32;
    const float* wr = ar + h * 32;
    float sl = 0.f, sr = 0.f;
#pragma unroll
    for (int d = 0; d < 32; ++d) {
        float v = f[d];
        sl = fmaf(v, wl[d], sl);
        sr = fmaf(v, wr[d], sr);
    }
    el[i] = sl;
    er[i] = sr;
}

// ---------------------------------------------------------------------------
// Generic fill
// ---------------------------------------------------------------------------
__global__ void fill_kernel(float* __restrict__ p, float v, int n) {
    int i = blockIdx.x * blockDim.x + threadIdx.x;
    if (i < n) p[i] = v;
}

__device__ __forceinline__ void atomicMaxFloat(float* addr, float val) {
    if (val >= 0.f) atomicMax((int*)addr, __float_as_int(val));
    else            atomicMin((unsigned int*)addr, __float_as_uint(val));
}

// ---------------------------------------------------------------------------
// Edge pass 1: segment max of leaky-relu'd scores into m[dst,h]
// ---------------------------------------------------------------------------
__global__ void edge_max_kernel(const int* __restrict__ src,
                                const int* __restrict__ dst,
                                const float* __restrict__ el,
                                const float* __restrict__ er,
                                float* __restrict__ m, int E) {
    int i = blockIdx.x * blockDim.x + threadIdx.x;   // over E*4
    if (i >= E * 4) return;
    int e = i >> 2, h = i & 3;
    int s = src[e], d = dst[e];
    float v = el[s * 4 + h] + er[d * 4 + h];
    v = v > 0.f ? v : NEG_SLOPE * v;
    atomicMaxFloat(&m[d * 4 + h], v);
}

// ---------------------------------------------------------------------------
// Edge pass 2: ex = exp(e - m[dst]); denom[dst] += ex
// ---------------------------------------------------------------------------
__global__ void edge_exp_kernel(const int* __restrict__ src,
                                const int* __restrict__ dst,
                                const float* __restrict__ el,
                                const float* __restrict__ er,
                                const float* __restrict__ m,
                                float* __restrict__ ex,
                                float* __restrict__ denom, int E) {
    int i = blockIdx.x * blockDim.x + threadIdx.x;   // over E*4
    if (i >= E * 4) return;
    int e = i >> 2, h = i & 3;
    int s = src[e], d = dst[e];
    float v = el[s * 4 + h] + er[d * 4 + h];
    v = v > 0.f ? v : NEG_SLOPE * v;
    float xv = __expf(v - m[d * 4 + h]);
    ex[i] = xv;
    atomicAdd(&denom[d * 4 + h], xv);
}

// ---------------------------------------------------------------------------
// Edge pass 3: o[dst, h, d] += alpha * feat[src, h, d]
// one thread per (edge, h*32+d); feat reads coalesced across the 128 lanes
// ---------------------------------------------------------------------------
__global__ void edge_agg_kernel(const int* __restrict__ src,
                                const int* __restrict__ dst,
                                const float* __restrict__ ex,
                                const float* __restrict__ denom,
                                const float* __restrict__ feat,
                                float* __restrict__ o, long long E) {
    long long t = (long long)blockIdx.x * blockDim.x + threadIdx.x; // E*128
    if (t >= E * 128) return;
    int e = (int)(t >> 7);
    int i = (int)(t & 127);
    int h = i >> 5;
    int s = src[e], d = dst[e];
    float alpha = ex[e * 4 + h] / fmaxf(denom[d * 4 + h], 1e-9f);
    atomicAdd(&o[d * 128 + i], alpha * feat[s * 128 + i]);
}

// ---------------------------------------------------------------------------
// Layer-1 combine: h1 += elu(o + b1[r])
// ---------------------------------------------------------------------------
__global__ void combine1_kernel(const float* __restrict__ o,
                                const float* __restrict__ b1r,
                                float* __restrict__ h1, int N) {
    int i = blockIdx.x * blockDim.x + threadIdx.x;   // over N*128
    if (i >= N * 128) return;
    float x = o[i] + b1r[i & 127];
    h1[i] += x > 0.f ? x : expm1f(x);
}

// ---------------------------------------------------------------------------
// Layer-2 combine: acc2 += o  (residual/bias folded into finalize)
// ---------------------------------------------------------------------------
__global__ void combine2_kernel(const float* __restrict__ o,
                                float* __restrict__ acc2, int N) {
    int i = blockIdx.x * blockDim.x + threadIdx.x;   // over N*128
    if (i >= N * 128) return;
    acc2[i] += o[i];
}

// ---------------------------------------------------------------------------
// Finalize: out[n,d] = mean_h(acc2 + R*h1 + sum_r b2[r])
// ---------------------------------------------------------------------------
__global__ void finalize_kernel(const float* __restrict__ acc2,
                                const float* __restrict__ h1,
                                const float* __restrict__ b2,
                                float* __restrict__ out, int N) {
    int i = blockIdx.x * blockDim.x + threadIdx.x;   // over N*32
    if (i >= N * 32) return;
    int n = i >> 5, d = i & 31;
    float s = 0.f;
#pragma unroll
    for (int h = 0; h < 4; ++h) {
        int j = h * 32 + d;
        s += acc2[n * 128 + j] + 2.0f * h1[n * 128 + j] + b2[j] + b2[128 + j];
    }
    out[i] = 0.25f * s;
}

// ===========================================================================
extern "C" void kernel_launch(void* const* d_in, const int* in_sizes, int n_in,
                              void* d_out, int out_size, void* d_ws, size_t ws_size,
                              hipStream_t stream) {
    const float* x   = (const float*)d_in[0];   // [N,128]
    const float* W1  = (const float*)d_in[1];   // [2,128,128]
    const float* al1 = (const float*)d_in[2];   // [2,4,32]
    const float* ar1 = (const float*)d_in[3];   // [2,4,32]
    const float* b1  = (const float*)d_in[4];   // [2,128]
    const float* W2  = (const float*)d_in[5];   // [2,128,128]
    const float* al2 = (const float*)d_in[6];   // [2,4,32]
    const float* ar2 = (const float*)d_in[7];   // [2,4,32]
    const float* b2  = (const float*)d_in[8];   // [2,128]
    const int* edges = (const int*)d_in[9];     // [2,2,E]
    float* out = (float*)d_out;                 // [N,32]

    const int N = in_sizes[0] / 128;            // 50000
    const int E = in_sizes[9] / 4;              // 800000
    const int R = 2;

    // workspace layout (floats)
    float* ws    = (float*)d_ws;
    float* feat  = ws;                      // N*128
    float* el    = feat  + (size_t)N * 128; // N*4
    float* er    = el    + (size_t)N * 4;
    float* m     = er    + (size_t)N * 4;
    float* denom = m     + (size_t)N * 4;
    float* ex    = denom + (size_t)N * 4;   // E*4
    float* o     = ex    + (size_t)E * 4;   // N*128
    float* h1    = o     + (size_t)N * 128; // N*128
    float* acc2  = h1    + (size_t)N * 128; // N*128

    const int TPB = 256;
    dim3 gemm_grid(N / 16, 2), gemm_block(128);
    const int nh    = N * 4;
    const int nfeat = N * 128;
    const int eh    = E * 4;
    const long long ew = (long long)E * 128;

    auto run_layer = [&](const float* hin, const float* W, const float* al,
                         const float* ar, int r, bool layer1) {
        gemm128_wmma_f32<<<gemm_grid, gemm_block, 0, stream>>>(
            hin, W + (size_t)r * 128 * 128, feat, N);
        score_kernel<<<(nh + TPB - 1) / TPB, TPB, 0, stream>>>(
            feat, al + (size_t)r * 128, ar + (size_t)r * 128, el, er, N);
        fill_kernel<<<(nh + TPB - 1) / TPB, TPB, 0, stream>>>(m, -INFINITY, nh);
        fill_kernel<<<(nh + TPB - 1) / TPB, TPB, 0, stream>>>(denom, 0.f, nh);
        fill_kernel<<<(nfeat + TPB - 1) / TPB, TPB, 0, stream>>>(o, 0.f, nfeat);
        const int* esrc = edges + (size_t)r * 2 * E;
        const int* edst = esrc + E;
        edge_max_kernel<<<(eh + TPB - 1) / TPB, TPB, 0, stream>>>(
            esrc, edst, el, er, m, E);
        edge_exp_kernel<<<(eh + TPB - 1) / TPB, TPB, 0, stream>>>(
            esrc, edst, el, er, m, ex, denom, E);
        edge_agg_kernel<<<(unsigned)((ew + TPB - 1) / TPB), TPB, 0, stream>>>(
            esrc, edst, ex, denom, feat, o, E);
        if (layer1) {
            combine1_kernel<<<(nfeat + TPB - 1) / TPB, TPB, 0, stream>>>(
                o, b1 + (size_t)r * 128, h1, N);
        } else {
            combine2_kernel<<<(nfeat + TPB - 1) / TPB, TPB, 0, stream>>>(o, acc2, N);
        }
    };

    // ---- layer 1 ----
    fill_kernel<<<(nfeat + TPB - 1) / TPB, TPB, 0, stream>>>(h1, 0.f, nfeat);
    for (int r = 0; r < R; ++r)
        run_layer(x, W1, al1, ar1, r, /*layer1=*/true);

    // ---- layer 2 ----
    fill_kernel<<<(nfeat + TPB - 1) / TPB, TPB, 0, stream>>>(acc2, 0.f, nfeat);
    for (int r = 0; r < R; ++r)
        run_layer(h1, W2, al2, ar2, r, /*layer1=*/false);

    // ---- finalize: mean over heads + residual + biases ----
    const int nout = N * 32;
    finalize_kernel<<<(nout + TPB - 1) / TPB, TPB, 0, stream>>>(
        acc2, h1, b2, out, N);
}